// AGDNConv_14173392077070
// MI455X (gfx1250) — compile-verified
//
#include <hip/hip_runtime.h>
#include <hip/hip_bf16.h>

#define N_NODES 100000
#define E_EDGES 1600000
#define D_IN    128
#define H_HEADS 2
#define F_OUT   64
#define HF      128      // H_HEADS * F_OUT
#define K_HOPS  3
#define SLOPE   0.2f

typedef __attribute__((ext_vector_type(2))) float v2f;
typedef __attribute__((ext_vector_type(8))) float v8f;

// ---------------------------------------------------------------------------
// sign-aware float atomic max (works for mixed-sign values, init = -inf)
// ---------------------------------------------------------------------------
__device__ __forceinline__ void atomicMaxF(float* addr, float val) {
    if (val >= 0.0f) {
        atomicMax((int*)addr, __float_as_int(val));
    } else {
        atomicMin((unsigned int*)addr, __float_as_uint(val));
    }
}

// ---------------------------------------------------------------------------
// generic fills
// ---------------------------------------------------------------------------
__global__ __launch_bounds__(256) void fill_kernel(float* __restrict__ p, float v, long n) {
    long i = (long)blockIdx.x * blockDim.x + threadIdx.x;
    if (i < n) p[i] = v;
}
__global__ __launch_bounds__(256) void fill_int_kernel(int* __restrict__ p, int v, long n) {
    long i = (long)blockIdx.x * blockDim.x + threadIdx.x;
    if (i < n) p[i] = v;
}

// ---------------------------------------------------------------------------
// Fused projection GEMM via V_WMMA_F32_16X16X4_F32.
//   h0  = x @ W_src                 [N, 128]
//   res = x @ W_dst + b_dst         [N, 128]
// One block = 16 rows of x. 8 waves; wave w owns output columns [16w,16w+16).
// ---------------------------------------------------------------------------
__global__ __launch_bounds__(256)
void gemm_proj_kernel(const float* __restrict__ x,
                      const float* __restrict__ Wsrc,
                      const float* __restrict__ Wdst,
                      const float* __restrict__ bdst,
                      float* __restrict__ h0,
                      float* __restrict__ res) {
    const int wave = threadIdx.x >> 5;
    const int lane = threadIdx.x & 31;
    const int row0 = blockIdx.x * 16;
    const int col0 = wave * 16;
    const int m    = lane & 15;           // row within tile for A
    const int kh   = (lane >> 4) * 2;     // K sub-offset (0 or 2)
    const int nn   = col0 + (lane & 15);  // output column for B / C / D

    v8f ch = {};
    v8f cr = {};

    const float* xrow = x + (size_t)(row0 + m) * D_IN;
    #pragma unroll 4
    for (int k = 0; k < D_IN; k += 4) {
        v2f a;
        a.x = xrow[k + kh];
        a.y = xrow[k + kh + 1];
        v2f bs;
        bs.x = Wsrc[(size_t)(k + kh) * HF + nn];
        bs.y = Wsrc[(size_t)(k + kh + 1) * HF + nn];
        ch = __builtin_amdgcn_wmma_f32_16x16x4_f32(false, a, false, bs, (short)0, ch, false, false);
        v2f bd;
        bd.x = Wdst[(size_t)(k + kh) * HF + nn];
        bd.y = Wdst[(size_t)(k + kh + 1) * HF + nn];
        cr = __builtin_amdgcn_wmma_f32_16x16x4_f32(false, a, false, bd, (short)0, cr, false, false);
    }

    const float bb = bdst[nn];
    #pragma unroll
    for (int r = 0; r < 8; ++r) {
        const int row = row0 + r + (lane >> 4) * 8;
        h0 [(size_t)row * HF + nn] = ch[r];
        res[(size_t)row * HF + nn] = cr[r] + bb;
    }
}

// ---------------------------------------------------------------------------
// attention projections: a_s = x@W_as, a_d = x@W_ad  (H=2 columns each)
// ---------------------------------------------------------------------------
__global__ __launch_bounds__(256)
void attn_proj_kernel(const float* __restrict__ x,
                      const float* __restrict__ Was,
                      const float* __restrict__ Wad,
                      float* __restrict__ a_s,
                      float* __restrict__ a_d) {
    const int n = blockIdx.x * blockDim.x + threadIdx.x;
    if (n >= N_NODES) return;
    const float* xr = x + (size_t)n * D_IN;
    float s0 = 0.f, s1 = 0.f, d0 = 0.f, d1 = 0.f;
    #pragma unroll 8
    for (int k = 0; k < D_IN; ++k) {
        const float xv = xr[k];
        s0 += xv * Was[k * 2 + 0];
        s1 += xv * Was[k * 2 + 1];
        d0 += xv * Wad[k * 2 + 0];
        d1 += xv * Wad[k * 2 + 1];
    }
    a_s[n * 2 + 0] = s0;  a_s[n * 2 + 1] = s1;
    a_d[n * 2 + 0] = d0;  a_d[n * 2 + 1] = d1;
}

// ---------------------------------------------------------------------------
// CSR build (by destination): degree count -> prefix scan -> edge-id scatter
// ---------------------------------------------------------------------------
__global__ __launch_bounds__(256)
void degree_kernel(const int* __restrict__ dst, int* __restrict__ deg) {
    const int e = blockIdx.x * blockDim.x + threadIdx.x;
    if (e < E_EDGES) atomicAdd(&deg[dst[e]], 1);
}

#define SCAN_THREADS 1024
__global__ __launch_bounds__(SCAN_THREADS)
void scan_kernel(const int* __restrict__ deg, int* __restrict__ rowptr, int n) {
    __shared__ int buf[SCAN_THREADS];
    __shared__ int carry_s;
    if (threadIdx.x == 0) carry_s = 0;
    __syncthreads();
    for (int base = 0; base < n; base += SCAN_THREADS) {
        const int i = base + (int)threadIdx.x;
        const int v = (i < n) ? deg[i] : 0;
        int x = v;
        buf[threadIdx.x] = x;
        __syncthreads();
        for (int offs = 1; offs < SCAN_THREADS; offs <<= 1) {
            const int y = (threadIdx.x >= (unsigned)offs) ? buf[threadIdx.x - offs] : 0;
            __syncthreads();
            x += y;
            buf[threadIdx.x] = x;
            __syncthreads();
        }
        const int carry = carry_s;
        if (i < n) rowptr[i] = carry + x - v;        // exclusive prefix
        __syncthreads();
        if (threadIdx.x == SCAN_THREADS - 1) carry_s = carry + x;
        __syncthreads();
    }
    if (threadIdx.x == 0) rowptr[n] = carry_s;       // == E_EDGES
}

__global__ __launch_bounds__(256)
void scatter_kernel(const int* __restrict__ dst, const int* __restrict__ rowptr,
                    int* __restrict__ cursor, int* __restrict__ perm) {
    const int e = blockIdx.x * blockDim.x + threadIdx.x;
    if (e >= E_EDGES) return;
    const int d = dst[e];
    const int p = atomicAdd(&cursor[d], 1);
    perm[rowptr[d] + p] = e;
}

// ---------------------------------------------------------------------------
// edge pass 1: e = leaky(a_s[src] + a_d[dst]); segment max by dst and by src
// ---------------------------------------------------------------------------
__global__ __launch_bounds__(256)
void edge_logit_max_kernel(const int* __restrict__ src, const int* __restrict__ dst,
                           const float* __restrict__ a_s, const float* __restrict__ a_d,
                           float* __restrict__ e_ws,
                           float* __restrict__ maxd, float* __restrict__ maxs) {
    const long t = (long)blockIdx.x * blockDim.x + threadIdx.x;
    if (t >= (long)E_EDGES * H_HEADS) return;
    const int e  = (int)(t >> 1);
    const int hd = (int)(t & 1);
    const int s = src[e], d = dst[e];
    float v = a_s[s * 2 + hd] + a_d[d * 2 + hd];
    v = (v >= 0.f) ? v : SLOPE * v;
    e_ws[t] = v;
    atomicMaxF(&maxd[d * 2 + hd], v);
    atomicMaxF(&maxs[s * 2 + hd], v);
}

// ---------------------------------------------------------------------------
// edge pass 2: segment sums of exp(e - max)
// ---------------------------------------------------------------------------
__global__ __launch_bounds__(256)
void edge_expsum_kernel(const int* __restrict__ src, const int* __restrict__ dst,
                        const float* __restrict__ e_ws,
                        const float* __restrict__ maxd, const float* __restrict__ maxs,
                        float* __restrict__ sumd, float* __restrict__ sums) {
    const long t = (long)blockIdx.x * blockDim.x + threadIdx.x;
    if (t >= (long)E_EDGES * H_HEADS) return;
    const int e  = (int)(t >> 1);
    const int hd = (int)(t & 1);
    const int s = src[e], d = dst[e];
    const float v = e_ws[t];
    atomicAdd(&sumd[d * 2 + hd], expf(v - maxd[d * 2 + hd]));
    atomicAdd(&sums[s * 2 + hd], expf(v - maxs[s * 2 + hd]));
}

// ---------------------------------------------------------------------------
// edge pass 3: a = sqrt(clip(softmax_dst,1e-9) * clip(softmax_src,1e-9))
// ---------------------------------------------------------------------------
__global__ __launch_bounds__(256)
void edge_coef_kernel(const int* __restrict__ src, const int* __restrict__ dst,
                      float* __restrict__ e_ws,
                      const float* __restrict__ maxd, const float* __restrict__ maxs,
                      const float* __restrict__ sumd, const float* __restrict__ sums) {
    const long t = (long)blockIdx.x * blockDim.x + threadIdx.x;
    if (t >= (long)E_EDGES * H_HEADS) return;
    const int e  = (int)(t >> 1);
    const int hd = (int)(t & 1);
    const int s = src[e], d = dst[e];
    const float v = e_ws[t];
    const float pd = expf(v - maxd[d * 2 + hd]) / sumd[d * 2 + hd];
    const float ps = expf(v - maxs[s * 2 + hd]) / sums[s * 2 + hd];
    e_ws[t] = sqrtf(fmaxf(pd, 1e-9f) * fmaxf(ps, 1e-9f));
}

// ---------------------------------------------------------------------------
// one diffusion hop, gather formulation (no atomics):
//   hout[d,:] = sum_{e in CSR row d} coef[e] * hin[src[e],:]
// one wave per destination node; lane owns a float4 column slice (head-aligned)
// ---------------------------------------------------------------------------
__global__ __launch_bounds__(256)
void spmm_gather_kernel(const int* __restrict__ src,
                        const int* __restrict__ rowptr, const int* __restrict__ perm,
                        const float* __restrict__ coef,
                        const float* __restrict__ hin, float* __restrict__ hout) {
    const int d = blockIdx.x * 8 + (threadIdx.x >> 5);
    if (d >= N_NODES) return;
    const int lane = threadIdx.x & 31;
    const int col  = lane * 4;
    const int head = col >> 6;
    const int beg = rowptr[d], end = rowptr[d + 1];

    float4 acc = {0.f, 0.f, 0.f, 0.f};
    for (int j = beg; j < end; ++j) {
        const int e = perm[j];
        if (j + 1 < end) {  // hide random-gather latency (global_prefetch_b8)
            const int e2 = perm[j + 1];
            __builtin_prefetch(hin + (size_t)src[e2] * HF + col, 0, 1);
        }
        const int   s = src[e];
        const float a = coef[(size_t)e * 2 + head];
        const float4 hv = *(const float4*)(hin + (size_t)s * HF + col);
        acc.x += hv.x * a;
        acc.y += hv.y * a;
        acc.z += hv.z * a;
        acc.w += hv.w * a;
    }
    *(float4*)(hout + (size_t)d * HF + col) = acc;
}

// ---------------------------------------------------------------------------
// feat_trans (per-head norm + scale/offset + pos_emb), hop attention softmax,
// weighted combine + residual. One wave per (node, head); 2 feats per lane.
// ---------------------------------------------------------------------------
__global__ __launch_bounds__(256)
void final_kernel(const float* __restrict__ hs0, const float* __restrict__ hs1,
                  const float* __restrict__ hs2, const float* __restrict__ res,
                  const float* __restrict__ scale, const float* __restrict__ offset,
                  const float* __restrict__ pos,
                  const float* __restrict__ attn_l, const float* __restrict__ attn_r,
                  float* __restrict__ out) {
    const int wid = blockIdx.x * 8 + (threadIdx.x >> 5);
    if (wid >= N_NODES * H_HEADS) return;
    const int lane = threadIdx.x & 31;
    const int n = wid >> 1, hd = wid & 1;
    const size_t base = (size_t)n * HF + hd * F_OUT;
    const int f0 = lane * 2;

    const float* hsk[K_HOPS] = { hs0, hs1, hs2 };
    float hv[K_HOPS][2];
    #pragma unroll
    for (int k = 0; k < K_HOPS; ++k) {
        const float v0 = hsk[k][base + f0];
        const float v1 = hsk[k][base + f0 + 1];
        float s  = v0 + v1;
        float sq = v0 * v0 + v1 * v1;
        #pragma unroll
        for (int m = 16; m; m >>= 1) {
            s  += __shfl_xor(s,  m, 32);
            sq += __shfl_xor(sq, m, 32);
        }
        const float mean = s * (1.0f / 64.0f);
        const float var  = sq * (1.0f / 64.0f) - mean * mean + 1e-9f;
        const float rstd = rsqrtf(var);
        const int si = k * HF + hd * F_OUT + f0;
        hv[k][0] = (v0 - mean) * scale[si]     * rstd + offset[si]     + pos[si];
        hv[k][1] = (v1 - mean) * scale[si + 1] * rstd + offset[si + 1] + pos[si + 1];
    }

    const int ai = hd * F_OUT + f0;
    float al = hv[0][0] * attn_l[ai] + hv[0][1] * attn_l[ai + 1];
    #pragma unroll
    for (int m = 16; m; m >>= 1) al += __shfl_xor(al, m, 32);

    float ar[K_HOPS];
    #pragma unroll
    for (int k = 0; k < K_HOPS; ++k) {
        float t = hv[k][0] * attn_r[ai] + hv[k][1] * attn_r[ai + 1];
        #pragma unroll
        for (int m = 16; m; m >>= 1) t += __shfl_xor(t, m, 32);
        t += al;
        ar[k] = (t >= 0.f) ? t : SLOPE * t;
    }
    const float mx = fmaxf(ar[0], fmaxf(ar[1], ar[2]));
    const float e0 = expf(ar[0] - mx), e1 = expf(ar[1] - mx), e2 = expf(ar[2] - mx);
    const float inv = 1.0f / (e0 + e1 + e2);
    const float w0 = e0 * inv, w1 = e1 * inv, w2 = e2 * inv;

    out[base + f0]     = hv[0][0] * w0 + hv[1][0] * w1 + hv[2][0] * w2 + res[base + f0];
    out[base + f0 + 1] = hv[0][1] * w0 + hv[1][1] * w1 + hv[2][1] * w2 + res[base + f0 + 1];
}

// ---------------------------------------------------------------------------
extern "C" void kernel_launch(void* const* d_in, const int* in_sizes, int n_in,
                              void* d_out, int out_size, void* d_ws, size_t ws_size,
                              hipStream_t stream) {
    const float* x     = (const float*)d_in[0];
    const int*   src   = (const int*)  d_in[1];
    const int*   dst   = (const int*)  d_in[2];
    const float* Wsrc  = (const float*)d_in[3];
    const float* Wdst  = (const float*)d_in[4];
    const float* bdst  = (const float*)d_in[5];
    const float* Was   = (const float*)d_in[6];
    const float* Wad   = (const float*)d_in[7];
    const float* scale = (const float*)d_in[8];
    const float* offs  = (const float*)d_in[9];
    const float* pos   = (const float*)d_in[10];
    const float* attl  = (const float*)d_in[11];
    const float* attr  = (const float*)d_in[12];
    float* out = (float*)d_out;
    (void)in_sizes; (void)n_in; (void)out_size; (void)ws_size;

    char* ws = (char*)d_ws;
    size_t off = 0;
    auto alloc = [&](size_t bytes) -> void* {
        void* p = (void*)(ws + off);
        off += (bytes + 255) & ~(size_t)255;
        return p;
    };
    const size_t NODE_FEATS = (size_t)N_NODES * HF * sizeof(float);
    float* h0     = (float*)alloc(NODE_FEATS);   // hop input; reused as hstack[2]
    float* hs0    = (float*)alloc(NODE_FEATS);   // hstack[0]
    float* hs1    = (float*)alloc(NODE_FEATS);   // hstack[1]
    float* res    = (float*)alloc(NODE_FEATS);
    float* a_s    = (float*)alloc((size_t)N_NODES * 2 * sizeof(float));
    float* a_d    = (float*)alloc((size_t)N_NODES * 2 * sizeof(float));
    float* maxd   = (float*)alloc((size_t)N_NODES * 2 * sizeof(float));
    float* maxs   = (float*)alloc((size_t)N_NODES * 2 * sizeof(float));
    float* sumd   = (float*)alloc((size_t)N_NODES * 2 * sizeof(float));
    float* sums   = (float*)alloc((size_t)N_NODES * 2 * sizeof(float));
    float* e_ws   = (float*)alloc((size_t)E_EDGES * 2 * sizeof(float));   // logits -> coefs
    int*   rowptr = (int*)  alloc((size_t)(N_NODES + 1) * sizeof(int));
    int*   deg    = (int*)  alloc((size_t)N_NODES * sizeof(int));         // also cursor
    int*   perm   = (int*)  alloc((size_t)E_EDGES * sizeof(int));

    const long nNH = (long)N_NODES * 2;
    const long nEH = (long)E_EDGES * H_HEADS;
    const float NEG_INF = -__builtin_huge_valf();

    const int gNH   = (int)((nNH + 255) / 256);
    const int gN    = (N_NODES + 255) / 256;
    const int gE    = (E_EDGES + 255) / 256;
    const int gEH   = (int)((nEH + 255) / 256);
    const int gWAVE = (N_NODES + 7) / 8;           // one wave per node

    // init
    fill_kernel<<<gNH, 256, 0, stream>>>(maxd, NEG_INF, nNH);
    fill_kernel<<<gNH, 256, 0, stream>>>(maxs, NEG_INF, nNH);
    fill_kernel<<<gNH, 256, 0, stream>>>(sumd, 0.0f, nNH);
    fill_kernel<<<gNH, 256, 0, stream>>>(sums, 0.0f, nNH);
    fill_int_kernel<<<gN, 256, 0, stream>>>(deg, 0, N_NODES);

    // dense projections (WMMA) + attention projections
    gemm_proj_kernel<<<N_NODES / 16, 256, 0, stream>>>(x, Wsrc, Wdst, bdst, h0, res);
    attn_proj_kernel<<<gN, 256, 0, stream>>>(x, Was, Wad, a_s, a_d);

    // CSR by destination (built once, reused for all K hops)
    degree_kernel<<<gE, 256, 0, stream>>>(dst, deg);
    scan_kernel<<<1, SCAN_THREADS, 0, stream>>>(deg, rowptr, N_NODES);
    fill_int_kernel<<<gN, 256, 0, stream>>>(deg, 0, N_NODES);   // reset as cursor
    scatter_kernel<<<gE, 256, 0, stream>>>(dst, rowptr, deg, perm);

    // dual-normalized edge attention
    edge_logit_max_kernel<<<gEH, 256, 0, stream>>>(src, dst, a_s, a_d, e_ws, maxd, maxs);
    edge_expsum_kernel   <<<gEH, 256, 0, stream>>>(src, dst, e_ws, maxd, maxs, sumd, sums);
    edge_coef_kernel     <<<gEH, 256, 0, stream>>>(src, dst, e_ws, maxd, maxs, sumd, sums);

    // K = 3 diffusion hops, pure gather (no atomics, no output zeroing)
    spmm_gather_kernel<<<gWAVE, 256, 0, stream>>>(src, rowptr, perm, e_ws, h0,  hs0);
    spmm_gather_kernel<<<gWAVE, 256, 0, stream>>>(src, rowptr, perm, e_ws, hs0, hs1);
    spmm_gather_kernel<<<gWAVE, 256, 0, stream>>>(src, rowptr, perm, e_ws, hs1, h0);

    // norm + hop attention + combine + residual
    final_kernel<<<(N_NODES * H_HEADS + 7) / 8, 256, 0, stream>>>(
        hs0, hs1, h0, res, scale, offs, pos, attl, attr, out);
}